// PairwiseInteractionModule_90460601188774
// MI455X (gfx1250) — compile-verified
//
#include <hip/hip_runtime.h>
#include <math.h>

// B=128, M=64, P=1024, A=21, D=64
// out[b,i,j] = mirrored upper-triangular pair-MLP scores, fp32.

typedef __attribute__((ext_vector_type(2))) float v2f;
typedef __attribute__((ext_vector_type(8))) float v8f;

#define LDS_STRIDE 68   // 4*n + k bank spread -> conflict-free b64 LDS reads

__device__ __forceinline__ float gelu_exact(float x) {
    // matches jax.nn.gelu(approximate=False) / torch default exact GELU
    return 0.5f * x * (1.0f + erff(x * 0.70710678118654752f));
}

// ---------------------------------------------------------------------------
// Kernel 1: E[b,m,:] = pos_table[clip(pos)] + aa_table[clip(aa)]
// one thread per element, 524288 total
// ---------------------------------------------------------------------------
__global__ __launch_bounds__(256)
void embed_kernel(const int* __restrict__ positions,
                  const int* __restrict__ amino,
                  const float* __restrict__ pos_table,
                  const float* __restrict__ aa_table,
                  float* __restrict__ E) {
    int t  = blockIdx.x * 256 + threadIdx.x;     // [0, B*M*D)
    int k  = t & 63;
    int bm = t >> 6;
    int p = positions[bm]; p = p < 0 ? 0 : (p > 1023 ? 1023 : p);
    int a = amino[bm];     a = a < 0 ? 0 : (a > 20   ? 20   : a);
    E[t] = pos_table[p * 64 + k] + aa_table[a * 64 + k];
}

// ---------------------------------------------------------------------------
// Kernel 2: one block per (batch, i-tile<=j-tile). 256 threads = 8 waves.
//   stage 1 (VALU): u = E_i @ W1[:64,:], v = E_j @ W1[64:,:]  -> LDS
//   stage 2 (WMMA): h1 = gelu(u_i + v_j + b1) kept in registers per lane,
//                   h2 = gelu(h1 @ W2 + b2) via v_wmma_f32_16x16x4_f32,
//                   s  = h2 @ W3 + b3 folded into accumulator reduction
//   stage 3: masked mirrored writes (i<j both orders, diagonal = 0)
// ---------------------------------------------------------------------------
__global__ __launch_bounds__(256)
void pair_kernel(const float* __restrict__ E,
                 const float* __restrict__ W1,
                 const float* __restrict__ b1,
                 const float* __restrict__ W2,
                 const float* __restrict__ b2,
                 const float* __restrict__ W3,
                 const float* __restrict__ b3,
                 float* __restrict__ out) {
    __shared__ float u_lds[16 * LDS_STRIDE];
    __shared__ float v_lds[16 * LDS_STRIDE];
    __shared__ float w2t [64 * LDS_STRIDE];    // w2t[n][k] = W2[k][n]
    __shared__ float b1s[64], b2s[64], w3s[64];

    const int tid = threadIdx.x;
    const int b   = blockIdx.y;
    const int tp  = blockIdx.x;                 // 0..9 enumerates it<=jt
    const int it  = (tp < 4) ? 0 : (tp < 7) ? 1 : (tp < 9) ? 2 : 3;
    const int start = it * 4 - (it * (it - 1)) / 2;  // 0,4,7,9
    const int jt  = tp - start + it;

    // ---- stage 1: per-token halves of layer 1 --------------------------------
    {
        const int n  = tid & 63;
        const int i0 = tid >> 6;               // 0..3
        const float* Ei = E + (b * 64 + it * 16) * 64;
        const float* Ej = E + (b * 64 + jt * 16) * 64;
        for (int s = 0; s < 4; ++s) {
            const int i = i0 + 4 * s;          // 0..15
            float su = 0.f, sv = 0.f;
            for (int k = 0; k < 64; ++k) {
                su = fmaf(Ei[i * 64 + k], W1[k * 64 + n],        su);
                sv = fmaf(Ej[i * 64 + k], W1[(64 + k) * 64 + n], sv);
            }
            u_lds[i * LDS_STRIDE + n] = su;
            v_lds[i * LDS_STRIDE + n] = sv;
        }
        for (int e = tid; e < 4096; e += 256) { // transpose W2 into LDS
            int k = e >> 6, nn = e & 63;
            w2t[nn * LDS_STRIDE + k] = W2[k * 64 + nn];
        }
        if (tid < 64) { b1s[tid] = b1[tid]; b2s[tid] = b2[tid]; w3s[tid] = W3[tid]; }
    }
    __syncthreads();

    const int lane = tid & 31;
    const int wv   = tid >> 5;                 // wave 0..7 -> i_local {2wv,2wv+1}
    const int jl   = lane & 15;                // A-row within tile == j_local
    const int half = lane >> 4;                // K sub-offset for 16x16x4 layout
    const int koff = 2 * half;

    // ---- h1 fragments in registers: A layout of V_WMMA_F32_16X16X4_F32 ------
    // lane<16:  A[M=lane][4kk+0], A[M=lane][4kk+1]
    // lane>=16: A[M=lane-16][4kk+2], A[M=lane-16][4kk+3]
    v2f hA[2][16];
    #pragma unroll
    for (int mt = 0; mt < 2; ++mt) {
        const int il = 2 * wv + mt;
        #pragma unroll
        for (int kk = 0; kk < 16; ++kk) {
            const int k0 = 4 * kk + koff;
            float x0 = u_lds[il * LDS_STRIDE + k0]     + v_lds[jl * LDS_STRIDE + k0]     + b1s[k0];
            float x1 = u_lds[il * LDS_STRIDE + k0 + 1] + v_lds[jl * LDS_STRIDE + k0 + 1] + b1s[k0 + 1];
            hA[mt][kk].x = gelu_exact(x0);
            hA[mt][kk].y = gelu_exact(x1);
        }
    }

    float p0[8], p1[8];
    #pragma unroll
    for (int r = 0; r < 8; ++r) { p0[r] = 0.f; p1[r] = 0.f; }

    // ---- layer 2 GEMM (WMMA) + fold layer 3 ---------------------------------
    #pragma unroll
    for (int nt = 0; nt < 4; ++nt) {
        const int n = nt * 16 + jl;
        v8f acc0 = {};
        v8f acc1 = {};
        #pragma unroll
        for (int kk = 0; kk < 16; ++kk) {
            const int k0 = 4 * kk + koff;
            // B layout: VGPR0 = rows K={0|2}, VGPR1 = rows K={1|3}, lane = column
            v2f bb = *(const v2f*)&w2t[n * LDS_STRIDE + k0];
            acc0 = __builtin_amdgcn_wmma_f32_16x16x4_f32(false, hA[0][kk], false, bb,
                                                         (short)0, acc0, false, false);
            acc1 = __builtin_amdgcn_wmma_f32_16x16x4_f32(false, hA[1][kk], false, bb,
                                                         (short)0, acc1, false, false);
        }
        const float b2n = b2s[n];
        const float w3n = w3s[n];
        #pragma unroll
        for (int r = 0; r < 8; ++r) {
            p0[r] = fmaf(gelu_exact(acc0[r] + b2n), w3n, p0[r]);
            p1[r] = fmaf(gelu_exact(acc1[r] + b2n), w3n, p1[r]);
        }
    }

    // ---- reduce over N (16 lanes per half) + masked mirrored store ----------
    const float bias3 = b3[0];
    #pragma unroll
    for (int mt = 0; mt < 2; ++mt) {
        const int ig = it * 16 + 2 * wv + mt;  // global i
        #pragma unroll
        for (int r = 0; r < 8; ++r) {
            float s = (mt == 0) ? p0[r] : p1[r];
            s += __shfl_xor(s, 1);
            s += __shfl_xor(s, 2);
            s += __shfl_xor(s, 4);
            s += __shfl_xor(s, 8);
            if (jl == 0) {                     // lanes 0 and 16 write their halves
                const int jg = jt * 16 + r + 8 * half;   // global j
                if (ig < jg) {
                    const float val = s + bias3;
                    out[(b * 64 + ig) * 64 + jg] = val;
                    out[(b * 64 + jg) * 64 + ig] = val;
                } else if (ig == jg) {
                    out[(b * 64 + ig) * 64 + jg] = 0.f;  // diagonal stays zero
                }
            }
        }
    }
}

// ---------------------------------------------------------------------------
extern "C" void kernel_launch(void* const* d_in, const int* in_sizes, int n_in,
                              void* d_out, int out_size, void* d_ws, size_t ws_size,
                              hipStream_t stream) {
    const int*   positions = (const int*)  d_in[0];
    const int*   amino     = (const int*)  d_in[1];
    const float* pos_table = (const float*)d_in[2];
    const float* aa_table  = (const float*)d_in[3];
    const float* W1        = (const float*)d_in[4];
    const float* b1        = (const float*)d_in[5];
    const float* W2        = (const float*)d_in[6];
    const float* b2        = (const float*)d_in[7];
    const float* W3        = (const float*)d_in[8];
    const float* b3        = (const float*)d_in[9];
    float* out = (float*)d_out;
    float* E   = (float*)d_ws;                 // 128*64*64 fp32 = 2 MB scratch

    embed_kernel<<<dim3((128 * 64 * 64) / 256), dim3(256), 0, stream>>>(
        positions, amino, pos_table, aa_table, E);
    pair_kernel<<<dim3(10, 128), dim3(256), 0, stream>>>(
        E, W1, b1, W2, b2, W3, b3, out);
}